// RGATLayer_58531814310078
// MI455X (gfx1250) — compile-verified
//
#include <hip/hip_runtime.h>

typedef __attribute__((ext_vector_type(2))) float v2f;
typedef __attribute__((ext_vector_type(8))) float v8f;

#define N_NODES   50000
#define E_EDGES   800000
#define IN_DIM    128
#define OUT_DIM   32
#define NUM_RELS  20
#define NUM_HEADS 5
#define SELF_DIM  (NUM_HEADS * OUT_DIM)      // 160
#define TOT_COLS  (OUT_DIM + SELF_DIM)       // 192
#define TILES_M   (N_NODES / 16)             // 3125
#define TILES_N   (TOT_COLS / 16)            // 12
#define TOTAL_TILES (TILES_M * TILES_N)      // 37500
#define WREL_SZ   (NUM_RELS * 2 * OUT_DIM * NUM_HEADS)  // 6400 floats = 25.6 KB

// ---------------------------------------------------------------------------
// Kernel 1: fused projection  y[n, 0:192] = h[n] @ [fc_w ; self_fc_w]^T
//   cols 0..31   -> z workspace (edge-phase operand)
//   cols 32..191 -> d_out (self-interaction term; also initializes d_out)
// One wave per 16x16 output tile, V_WMMA_F32_16X16X4_F32 over K=128.
// ---------------------------------------------------------------------------
__global__ __launch_bounds__(256) void rgat_proj_wmma(
    const float* __restrict__ h,
    const float* __restrict__ fc_w,
    const float* __restrict__ self_fc_w,
    float* __restrict__ z,
    float* __restrict__ out)
{
    const int wave = blockIdx.x * (blockDim.x >> 5) + (threadIdx.x >> 5);
    if (wave >= TOTAL_TILES) return;          // wave-uniform: EXEC stays all-ones

    const int tm   = wave / TILES_N;
    const int tn   = wave - tm * TILES_N;
    const int lane = threadIdx.x & 31;
    const int mr   = lane & 15;               // A row (M) == B column (N) owner
    const int kk   = (lane >> 4) << 1;        // lanes 0-15: K+{0,1}; 16-31: K+{2,3}

    // Select which weight matrix this 16-col tile lives in (32 is tile-aligned)
    const float* wmat;
    int col0;
    if (tn < 2) { wmat = fc_w;      col0 = tn * 16;      }
    else        { wmat = self_fc_w; col0 = tn * 16 - 32; }

    const float* arow = h    + (size_t)(tm * 16 + mr) * IN_DIM + kk;
    const float* brow = wmat + (size_t)(col0   + mr) * IN_DIM + kk;

    v8f acc = {};
    #pragma unroll 4
    for (int k = 0; k < IN_DIM; k += 4) {
        v2f a = *(const v2f*)(arow + k);      // A[m][k+kk], A[m][k+kk+1]
        v2f b = *(const v2f*)(brow + k);      // B[k+kk][n] = W[n][k+kk], ...
        acc = __builtin_amdgcn_wmma_f32_16x16x4_f32(
            /*neg_a=*/false, a, /*neg_b=*/false, b,
            /*c_mod=*/(short)0, acc, /*reuse_a=*/false, /*reuse_b=*/false);
    }

    // C/D layout: VGPR r, lanes 0-15 -> row r; lanes 16-31 -> row r+8; col = lane&15
    const int rowbase = tm * 16 + ((lane >> 4) << 3);
    if (tn < 2) {
        #pragma unroll
        for (int r = 0; r < 8; ++r)
            z[(size_t)(rowbase + r) * OUT_DIM + col0 + mr] = acc[r];
    } else {
        #pragma unroll
        for (int r = 0; r < 8; ++r)
            out[(size_t)(rowbase + r) * SELF_DIM + col0 + mr] = acc[r];
    }
}

// ---------------------------------------------------------------------------
// Kernel 2: per-edge attention + scatter-add aggregation.
// One wave32 per edge; lane j owns channel j of OUT_DIM=32.
// Relation weights staged in LDS (25.6 KB, conflict-free: stride 5 vs 64 banks).
// E = 800000 = 100000 blocks * 8 waves exactly -> no guards around barrier.
// ---------------------------------------------------------------------------
__global__ __launch_bounds__(256) void rgat_edge(
    const float* __restrict__ z,
    const float* __restrict__ weight,
    const int*   __restrict__ src,
    const int*   __restrict__ dst,
    const int*   __restrict__ rel,
    float*       __restrict__ out)
{
    __shared__ float wlds[WREL_SZ];
    for (int i = threadIdx.x; i < WREL_SZ; i += blockDim.x)
        wlds[i] = weight[i];
    __syncthreads();

    const int e = blockIdx.x * (blockDim.x >> 5) + (threadIdx.x >> 5);
    const int j = threadIdx.x & 31;

    const int s = src[e];
    const int d = dst[e];
    const int r = rel[e];

    const float zs = z[(size_t)s * OUT_DIM + j];   // 128B contiguous per wave
    const float zd = z[(size_t)d * OUT_DIM + j];

    // weight[r][c][h] at r*320 + c*5 + h ; c=j (src half), c=32+j (dst half)
    const int base = r * (2 * OUT_DIM * NUM_HEADS) + j * NUM_HEADS;

    float a[NUM_HEADS];
    #pragma unroll
    for (int hh = 0; hh < NUM_HEADS; ++hh) {
        float p = zs * wlds[base + hh] + zd * wlds[base + OUT_DIM * NUM_HEADS + hh];
        // wave32 butterfly reduction -> every lane holds the full sum
        p += __shfl_xor(p, 16, 32);
        p += __shfl_xor(p, 8, 32);
        p += __shfl_xor(p, 4, 32);
        p += __shfl_xor(p, 2, 32);
        p += __shfl_xor(p, 1, 32);
        a[hh] = (p > 0.0f) ? p : 0.01f * p;        // leaky_relu(0.01)
    }

    float* orow = out + (size_t)d * SELF_DIM + j;
    #pragma unroll
    for (int hh = 0; hh < NUM_HEADS; ++hh)
        atomicAdd(orow + hh * OUT_DIM, a[hh] * zs); // L2-resident atomic scatter
}

// ---------------------------------------------------------------------------
extern "C" void kernel_launch(void* const* d_in, const int* in_sizes, int n_in,
                              void* d_out, int out_size, void* d_ws, size_t ws_size,
                              hipStream_t stream)
{
    const float* h         = (const float*)d_in[0];
    const float* fc_w      = (const float*)d_in[1];
    const float* self_fc_w = (const float*)d_in[2];
    const float* weight    = (const float*)d_in[3];
    const int*   src       = (const int*)d_in[4];
    const int*   dst       = (const int*)d_in[5];
    const int*   rel       = (const int*)d_in[6];

    float* out = (float*)d_out;
    float* z   = (float*)d_ws;                 // N*32 floats = 6.4 MB scratch

    const int blocks1 = (TOTAL_TILES + 7) / 8; // 8 waves / 256-thread block
    rgat_proj_wmma<<<blocks1, 256, 0, stream>>>(h, fc_w, self_fc_w, z, out);

    const int blocks2 = E_EDGES / 8;           // exactly 100000
    rgat_edge<<<blocks2, 256, 0, stream>>>(z, weight, src, dst, rel, out);
}